// SlayerNet_55241869361817
// MI455X (gfx1250) — compile-verified
//
#include <hip/hip_runtime.h>

typedef float v2f __attribute__((ext_vector_type(2)));
typedef float v8f __attribute__((ext_vector_type(8)));

#define THETA 10.0f

// ---------------------------------------------------------------------------
// Kernel 1: y[32xT] = w1[MxK] @ x[KxT] using V_WMMA_F32_16X16X4_F32.
// y is padded to 32 rows so the store epilogue is unconditional.
// CT_T > 0  => T is compile-time (immediate load/store offsets).
// ---------------------------------------------------------------------------
template <int CT_T>
__global__ void gemm_w1x_wmma(const float* __restrict__ w1,
                              const float* __restrict__ x,
                              float* __restrict__ y,
                              int M, int Kdim, int rtT) {
    const int T = (CT_T > 0) ? CT_T : rtT;

    constexpr int AS_STRIDE = 258;   // even (8B-aligned pairs), 258%64=2 (no bank conflicts)
    __shared__ float As[32 * AS_STRIDE];

    const int tid = threadIdx.x;
    // Cooperative zero-padded fill, branch-free: clamped load + select.
    for (int idx = tid; idx < 32 * 256; idx += 128) {
        const int row = idx >> 8;
        const int k   = idx & 255;
        const int rc  = (row < M) ? row : (M - 1);
        const int kc  = (k < Kdim) ? k : (Kdim - 1);
        const float v = w1[rc * Kdim + kc];
        As[row * AS_STRIDE + k] = (row < M && k < Kdim) ? v : 0.0f;
    }
    __syncthreads();

    const int wave = tid >> 5;
    const int lane = tid & 31;
    const int n0   = (blockIdx.x * 4 + wave) * 16;
    const int lm   = lane & 15;
    const bool hi  = lane >= 16;

    const int n    = n0 + lm;
    const int kmax = Kdim - 1;
    const int Kmain = Kdim & ~7;            // 248: all indices k0..k0+7 in-bounds
    const int Kpad  = (Kdim + 7) & ~7;      // 256

    v8f c0 = {};   // rows 0..15
    v8f c1 = {};   // rows 16..31 (rows >= M are dead padding)

    const float* As0 = &As[lm * AS_STRIDE];
    const float* As1 = &As[(16 + lm) * AS_STRIDE];

    // Per-lane B base: column n, row (hi ? 2 : 0). Offsets are 0,T,4T,5T (+8T/iter).
    const float* pB = x + (size_t)(hi ? 2 : 0) * T + n;

    for (int k0 = 0; k0 < Kmain; k0 += 8) {
        const int kA0 = k0 + (hi ? 2 : 0);
        const int kA1 = kA0 + 4;
        v2f a0, a1, a2, a3, b0, b1;

        a0.x = As0[kA0];  a0.y = As0[kA0 + 1];
        a1.x = As1[kA0];  a1.y = As1[kA0 + 1];
        a2.x = As0[kA1];  a2.y = As0[kA1 + 1];
        a3.x = As1[kA1];  a3.y = As1[kA1 + 1];

        b0.x = pB[0];
        b0.y = pB[(size_t)T];
        b1.x = pB[(size_t)4 * T];
        b1.y = pB[(size_t)5 * T];

        c0 = __builtin_amdgcn_wmma_f32_16x16x4_f32(false, a0, false, b0, (short)0, c0, false, false);
        c1 = __builtin_amdgcn_wmma_f32_16x16x4_f32(false, a1, false, b0, (short)0, c1, false, false);
        c0 = __builtin_amdgcn_wmma_f32_16x16x4_f32(false, a2, false, b1, (short)0, c0, false, false);
        c1 = __builtin_amdgcn_wmma_f32_16x16x4_f32(false, a3, false, b1, (short)0, c1, false, false);

        pB += (size_t)8 * T;
    }

    // Clamped tail (A is exact zero for k >= Kdim, so clamped B rows contribute 0).
    for (int k0 = Kmain; k0 < Kpad; k0 += 4) {
        const int kA = k0 + (hi ? 2 : 0);
        const int r0 = (kA     < kmax) ? kA     : kmax;
        const int r1 = (kA + 1 < kmax) ? kA + 1 : kmax;
        v2f a0, a1, b;
        a0.x = As0[kA];  a0.y = As0[kA + 1];
        a1.x = As1[kA];  a1.y = As1[kA + 1];
        b.x = x[(size_t)r0 * T + n];
        b.y = x[(size_t)r1 * T + n];
        c0 = __builtin_amdgcn_wmma_f32_16x16x4_f32(false, a0, false, b, (short)0, c0, false, false);
        c1 = __builtin_amdgcn_wmma_f32_16x16x4_f32(false, a1, false, b, (short)0, c1, false, false);
    }

    // D layout: VGPR r -> M = r (lanes 0-15) / 8+r (lanes 16-31); N = lane&15.
    // Unconditional stores into the 32-row padded output.
    float* p = y + (size_t)(hi ? 8 : 0) * T + n;
#pragma unroll
    for (int r = 0; r < 8; ++r)
        p[(size_t)r * T] = c0[r];
    float* q = p + (size_t)16 * T;
#pragma unroll
    for (int r = 0; r < 8; ++r)
        q[(size_t)r * T] = c1[r];
}

// ---------------------------------------------------------------------------
// Kernel 2/5: per-channel FIR, out[c,t] = sum_j srm[j] * z[c, t - K/2 + j].
// LDS-tiled: 256 outputs per block, window = 256 + K - 1 (<= 512 for K=257).
// ---------------------------------------------------------------------------
__global__ void fir_kernel(const float* __restrict__ z,
                           const float* __restrict__ srm,
                           float* __restrict__ out,
                           int T, int K) {
    __shared__ float tile[768];
    __shared__ float taps[320];

    const int c   = blockIdx.y;
    const int t0  = blockIdx.x * 256;
    const int tid = threadIdx.x;
    const int p   = K >> 1;
    const int W   = 256 + K - 1;

    const float* zc = z + (size_t)c * T;
    for (int i = tid; i < W; i += 256) {
        const int t = t0 - p + i;
        tile[i] = (t >= 0 && t < T) ? zc[t] : 0.0f;
    }
    for (int i = tid; i < K; i += 256) taps[i] = srm[i];
    __syncthreads();

    float acc = 0.0f;
#pragma unroll 4
    for (int j = 0; j < K; ++j) acc = fmaf(taps[j], tile[tid + j], acc);

    const int t = t0 + tid;
    if (t < T) out[(size_t)c * T + t] = acc;
}

// ---------------------------------------------------------------------------
// Kernel 3/6: sequential spike generation. One lane per neuron; 16-deep
// refractory state in registers. t unrolled x4 with b128 loads/stores to cut
// load-latency stalls on the serial chain. wave32, 1 block.
// ---------------------------------------------------------------------------
__global__ void scan_spikes(const float* __restrict__ pot,
                            const float* __restrict__ ref,
                            float* __restrict__ s_out,
                            int C, int T, int Lref) {
    const int c = threadIdx.x;
    if (c >= C) return;

    float refn[16];
#pragma unroll
    for (int j = 0; j < 16; ++j) refn[j] = (j + 1 < Lref) ? ref[j + 1] : 0.0f;

    float buf[16];
#pragma unroll
    for (int j = 0; j < 16; ++j) buf[j] = 0.0f;

    const float* pc = pot + (size_t)c * T;
    float*       so = s_out + (size_t)c * T;

    for (int t = 0; t < T; t += 4) {
        const float4 pv = *(const float4*)(pc + t);
        float pe[4] = {pv.x, pv.y, pv.z, pv.w};
        float se[4];
#pragma unroll
        for (int i = 0; i < 4; ++i) {
            const float u = pe[i] + buf[0];
            const float s = (u >= THETA) ? 1.0f : 0.0f;
#pragma unroll
            for (int j = 0; j < 15; ++j) buf[j] = buf[j + 1] + s * refn[j];
            buf[15] = s * refn[15];
            se[i] = s;
        }
        *(float4*)(so + t) = make_float4(se[0], se[1], se[2], se[3]);
    }
}

// ---------------------------------------------------------------------------
// Kernel 4: y2[t] = sum_c w2[c] * s1[c,t]; 4 columns per thread (b128).
// ---------------------------------------------------------------------------
__global__ void dot_w2(const float* __restrict__ s1,
                       const float* __restrict__ w2,
                       float* __restrict__ y2,
                       int C, int T) {
    const int t0 = (blockIdx.x * blockDim.x + threadIdx.x) * 4;
    if (t0 >= T) return;
    float ax = 0.0f, ay = 0.0f, az = 0.0f, aw = 0.0f;
    for (int c = 0; c < C; ++c) {
        const float  w = w2[c];
        const float4 v = *(const float4*)(s1 + (size_t)c * T + t0);
        ax = fmaf(w, v.x, ax);
        ay = fmaf(w, v.y, ay);
        az = fmaf(w, v.z, az);
        aw = fmaf(w, v.w, aw);
    }
    *(float4*)(y2 + t0) = make_float4(ax, ay, az, aw);
}

// ---------------------------------------------------------------------------
extern "C" void kernel_launch(void* const* d_in, const int* in_sizes, int n_in,
                              void* d_out, int out_size, void* d_ws, size_t ws_size,
                              hipStream_t stream) {
    const float* x   = (const float*)d_in[0];   // (1,1,1,250,T) f32
    const float* srm = (const float*)d_in[1];   // K = 257
    const float* ref = (const float*)d_in[2];   // Lref = 16
    const float* w1  = (const float*)d_in[3];   // (25,250)
    const float* w2  = (const float*)d_in[4];   // (1,25)
    float* out = (float*)d_out;                 // (T,) f32

    const int T    = out_size;            // 65536
    const int Ksrm = in_sizes[1];         // 257
    const int Lref = in_sizes[2];         // 16
    const int C    = in_sizes[0] / T;     // 250
    const int H    = in_sizes[3] / C;     // 25

    // Workspace layout (floats): y1[32*T] (padded) | h1[H*T] | s1[H*T] | y2[T] | h2[T]
    float* ws = (float*)d_ws;
    float* y1 = ws;
    float* h1 = y1 + (size_t)32 * T;
    float* s1 = h1 + (size_t)H * T;
    float* y2 = s1 + (size_t)H * T;
    float* h2 = y2 + (size_t)T;

    // 1) y1 = w1 @ x          (f32 WMMA; FIR commutes with channel mix)
    if (T == 65536)
        gemm_w1x_wmma<65536><<<T / 64, 128, 0, stream>>>(w1, x, y1, H, C, T);
    else
        gemm_w1x_wmma<0><<<T / 64, 128, 0, stream>>>(w1, x, y1, H, C, T);

    // 2) h1 = FIR(y1)
    {
        dim3 g(T / 256, H);
        fir_kernel<<<g, 256, 0, stream>>>(y1, srm, h1, T, Ksrm);
    }

    // 3) s1 = gen_spikes(h1)  (serial in t, parallel over 25 neurons)
    scan_spikes<<<1, 32, 0, stream>>>(h1, ref, s1, H, T, Lref);

    // 4) y2 = w2 @ s1
    dot_w2<<<T / 1024, 256, 0, stream>>>(s1, w2, y2, H, T);

    // 5) h2 = FIR(y2)
    {
        dim3 g(T / 256, 1);
        fir_kernel<<<g, 256, 0, stream>>>(y2, srm, h2, T, Ksrm);
    }

    // 6) out = gen_spikes(h2)
    scan_spikes<<<1, 32, 0, stream>>>(h2, ref, out, 1, T, Lref);
}